// ManoChamferSDFOptimizer_2611340116486
// MI455X (gfx1250) — compile-verified
//
#include <hip/hip_runtime.h>
#include <math.h>

typedef __attribute__((ext_vector_type(2))) float v2f;
typedef __attribute__((ext_vector_type(8))) float v8f;

#define EPSF 1e-6f

// ---------- order-preserving float<->uint encoding for atomic min/max ----------
__device__ __forceinline__ unsigned enc_f(float f) {
  unsigned u = __float_as_uint(f);
  return ((int)u < 0) ? ~u : (u ^ 0x80000000u);
}
__device__ __forceinline__ float dec_f(unsigned e) {
  unsigned u = ((int)e < 0) ? (e ^ 0x80000000u) : ~e;
  return __uint_as_float(u);
}

// ---------- shared small-vector math: rot matrix + world tips/mids ----------
__device__ void compute_hand(const float* palm_pos, const float* palm_rot, const float* qpos,
                             const float* tip_base, const float* mid_base,
                             const float* J_tip, const float* J_mid,
                             float R[9], float tips[5][3], float mids[5][3]) {
  float qx = palm_rot[0], qy = palm_rot[1], qz = palm_rot[2], qw = palm_rot[3];
  float qn = sqrtf(qx*qx + qy*qy + qz*qz + qw*qw);
  float x = qx/qn, y = qy/qn, z = qz/qn, w = qw/qn;
  R[0]=1.f-2.f*(y*y+z*z); R[1]=2.f*(x*y-z*w);     R[2]=2.f*(x*z+y*w);
  R[3]=2.f*(x*y+z*w);     R[4]=1.f-2.f*(x*x+z*z); R[5]=2.f*(y*z-x*w);
  R[6]=2.f*(x*z-y*w);     R[7]=2.f*(y*z+x*w);     R[8]=1.f-2.f*(x*x+y*y);
  for (int f = 0; f < 5; ++f) {
    float lt[3], lm[3];
    for (int d = 0; d < 3; ++d) {
      float t = tip_base[f*3+d], m = mid_base[f*3+d];
      for (int q = 0; q < 20; ++q) {
        t += J_tip[(f*3+d)*20+q] * qpos[q];
        m += J_mid[(f*3+d)*20+q] * qpos[q];
      }
      lt[d] = t; lm[d] = m;
    }
    for (int r = 0; r < 3; ++r) {
      tips[f][r] = R[r*3+0]*lt[0] + R[r*3+1]*lt[1] + R[r*3+2]*lt[2] + palm_pos[r];
      mids[f][r] = R[r*3+0]*lm[0] + R[r*3+1]*lm[1] + R[r*3+2]*lm[2] + palm_pos[r];
    }
  }
}

// ---------- kernel A: build WMMA A-matrix / C-accumulator constants, init reductions ----------
__global__ void prep_kernel(const float* hout, const float* palm_pos, const float* palm_rot,
                            const float* qpos, const float* tip_base, const float* mid_base,
                            const float* J_tip, const float* J_mid,
                            float* wsf, unsigned* red) {
  if (threadIdx.x != 0 || blockIdx.x != 0) return;
  float R[9], tips[5][3], mids[5][3];
  compute_hand(palm_pos, palm_rot, qpos, tip_base, mid_base, J_tip, J_mid, R, tips, mids);
  float n1 = sqrtf(hout[0]*hout[0] + hout[1]*hout[1] + hout[2]*hout[2]) + EPSF;
  float ho0 = hout[0]/n1, ho1 = hout[1]/n1, ho2 = hout[2]/n1;
  // queries 0..4 = tips, 5..9 = mids, 10..15 = zero pad
  for (int m = 0; m < 16; ++m) {
    float qx = 0.f, qy = 0.f, qz = 0.f;
    if (m < 5)       { qx = tips[m][0];   qy = tips[m][1];   qz = tips[m][2];   }
    else if (m < 10) { qx = mids[m-5][0]; qy = mids[m-5][1]; qz = mids[m-5][2]; }
    wsf[m]      = -2.f*qx;   // A row, K=0
    wsf[16+m]   = -2.f*qy;   // A row, K=1
    wsf[32+m]   = -2.f*qz;   // A row, K=2  (K=3 is zero)
    wsf[48+m]   = qx*qx + qy*qy + qz*qz;   // C accumulator = |q|^2
  }
  wsf[64] = ho0; wsf[65] = ho1; wsf[66] = ho2;
  for (int i = 0; i < 16; ++i) red[i] = 0x7F800000u;           // +inf bits (nonneg-min)
  red[16] = enc_f(__uint_as_float(0x7F800000u));               // proj min cell <- +inf
  red[17] = enc_f(__uint_as_float(0xFF800000u));               // proj max cell <- -inf
}

// ---------- kernel B: streaming chamfer + projection reduction via V_WMMA_F32_16X16X4_F32 ----------
// Each wave processes 32 consecutive points per iteration (two 16-point WMMA tiles,
// 384 contiguous bytes). Loop trip count is forced into SGPRs (readfirstlane) so loop
// control is pure SALU and EXEC stays all-ones across every WMMA, as the ISA requires.
__global__ __launch_bounds__(256) void chamfer_kernel(const float* __restrict__ pc,
                                                      const float* __restrict__ wsf,
                                                      unsigned* __restrict__ red,
                                                      int n_pairs) {
  const int lane = threadIdx.x & 31;
  const int h    = lane & 15;
  const bool lo  = lane < 16;
  const int wid  = __builtin_amdgcn_readfirstlane(
                       blockIdx.x * (blockDim.x >> 5) + (threadIdx.x >> 5));
  const int nw   = gridDim.x * (blockDim.x >> 5);

  // A matrix 16x4 f32: VGPR0 = {K0 | K2}, VGPR1 = {K1 | K3=0}  (rows = queries, pre-scaled by -2)
  v2f a;
  a.x = lo ? wsf[h] : wsf[32 + h];
  a.y = lo ? wsf[16 + h] : 0.f;
  // C 16x16 f32: VGPR v -> row (v | v+8), value = |q|^2 broadcast across columns
  v8f cacc;
#pragma unroll
  for (int v = 0; v < 8; ++v) cacc[v] = wsf[48 + (lo ? v : v + 8)];
  const float ox = wsf[64], oy = wsf[65], oz = wsf[66];

  float m[8];
#pragma unroll
  for (int v = 0; v < 8; ++v) m[v] = __uint_as_float(0x7F800000u);
  float pmin = __uint_as_float(0x7F800000u);
  float pmax = __uint_as_float(0xFF800000u);

  const size_t lane_off = (size_t)(3 * h);
  for (int g = wid; g < n_pairs; g += nw) {           // SALU loop, EXEC untouched
    const float* p = pc + (size_t)g * 96 + lane_off;
    // prefetch next grid-stride chunk (speculative, no counter)
    __builtin_prefetch(p + (size_t)nw * 96, 0, 1);
    float x0 = p[0],  y0 = p[1],  z0 = p[2];          // points 32g+h
    float x1 = p[48], y1 = p[49], z1 = p[50];         // points 32g+16+h
    // B matrix 4x16 f32: VGPR0 = {K0=x | K2=z}, VGPR1 = {K1=y | K3=0}
    v2f b0; b0.x = lo ? x0 : z0; b0.y = lo ? y0 : 0.f;
    v2f b1; b1.x = lo ? x1 : z1; b1.y = lo ? y1 : 0.f;
    // D[q,n] = |t_q|^2 - 2 t_q . p_n   (exact fp32 tensor op)
    v8f d0 = __builtin_amdgcn_wmma_f32_16x16x4_f32(false, a, false, b0, (short)0, cacc,
                                                   false, false);
    v8f d1 = __builtin_amdgcn_wmma_f32_16x16x4_f32(false, a, false, b1, (short)0, cacc,
                                                   false, false);
    float pn0 = fmaf(x0, x0, fmaf(y0, y0, z0 * z0));
    float pn1 = fmaf(x1, x1, fmaf(y1, y1, z1 * z1));
#pragma unroll
    for (int v = 0; v < 8; ++v) {
      m[v] = fminf(m[v], fmaxf(d0[v] + pn0, 0.f));
      m[v] = fminf(m[v], fmaxf(d1[v] + pn1, 0.f));
    }
    float pr0 = fmaf(x0, ox, fmaf(y0, oy, z0 * oz));
    float pr1 = fmaf(x1, ox, fmaf(y1, oy, z1 * oz));
    pmin = fminf(pmin, fminf(pr0, pr1));
    pmax = fmaxf(pmax, fmaxf(pr0, pr1));
  }

  // min across each 16-lane half (D rows 0-7 live in lanes 0-15, rows 8-15 in lanes 16-31)
#pragma unroll
  for (int s = 1; s < 16; s <<= 1) {
#pragma unroll
    for (int v = 0; v < 8; ++v) m[v] = fminf(m[v], __shfl_xor(m[v], s, 32));
  }
#pragma unroll
  for (int s = 1; s < 32; s <<= 1) {
    pmin = fminf(pmin, __shfl_xor(pmin, s, 32));
    pmax = fmaxf(pmax, __shfl_xor(pmax, s, 32));
  }

  if (lane == 0) {
#pragma unroll
    for (int v = 0; v < 8; ++v) atomicMin(&red[v], __float_as_uint(m[v]));  // queries 0..7
    atomicMin(&red[16], enc_f(pmin));
    atomicMax(&red[17], enc_f(pmax));
  } else if (lane == 16) {
    atomicMin(&red[8], __float_as_uint(m[0]));                              // query 8
    atomicMin(&red[9], __float_as_uint(m[1]));                              // query 9
  }
}

// ---------- kernel C: tail points + all scalar loss terms ----------
__global__ void finish_kernel(const float* pc, long n_points, long tail_start,
                              const float* center, const float* hout, const float* hlong,
                              const float* palm_pos, const float* palm_rot, const float* qpos,
                              const float* tip_base, const float* mid_base,
                              const float* J_tip, const float* J_mid,
                              const float* patch_local, const int* step_p,
                              const unsigned* red, float* out) {
  if (threadIdx.x != 0 || blockIdx.x != 0) return;
  float R[9], tips[5][3], mids[5][3];
  compute_hand(palm_pos, palm_rot, qpos, tip_base, mid_base, J_tip, J_mid, R, tips, mids);

  float ho[3], hl[3];
  {
    float n1 = sqrtf(hout[0]*hout[0] + hout[1]*hout[1] + hout[2]*hout[2]) + EPSF;
    float n2 = sqrtf(hlong[0]*hlong[0] + hlong[1]*hlong[1] + hlong[2]*hlong[2]) + EPSF;
    for (int i = 0; i < 3; ++i) { ho[i] = hout[i]/n1; hl[i] = hlong[i]/n2; }
  }
  float c0 = center[0]*ho[0] + center[1]*ho[1] + center[2]*ho[2];

  float dmin[10];
  for (int i = 0; i < 10; ++i) dmin[i] = __uint_as_float(red[i]);
  float rawmin = dec_f(red[16]);
  float rawmax = dec_f(red[17]);

  float Q[10][3], Q2[10];
  for (int f = 0; f < 5; ++f)
    for (int d = 0; d < 3; ++d) { Q[f][d] = tips[f][d]; Q[5+f][d] = mids[f][d]; }
  for (int i = 0; i < 10; ++i) Q2[i] = Q[i][0]*Q[i][0] + Q[i][1]*Q[i][1] + Q[i][2]*Q[i][2];

  for (long j = tail_start; j < n_points; ++j) {      // tail (<32 points)
    float x = pc[3*j], y = pc[3*j+1], z = pc[3*j+2];
    float pn2 = x*x + y*y + z*z;
    for (int i = 0; i < 10; ++i) {
      float d2 = pn2 - 2.f*(x*Q[i][0] + y*Q[i][1] + z*Q[i][2]) + Q2[i];
      dmin[i] = fminf(dmin[i], fmaxf(d2, 0.f));
    }
    float pr = x*ho[0] + y*ho[1] + z*ho[2];
    rawmin = fminf(rawmin, pr);
    rawmax = fmaxf(rawmax, pr);
  }

  float per_tip[5], per_mid[5];
  for (int f = 0; f < 5; ++f) { per_tip[f] = sqrtf(dmin[f]); per_mid[f] = sqrtf(dmin[5+f]); }
  float min_proj = rawmin - c0, max_proj = rawmax - c0;

  // ---- loss_align ----
  float vw[3] = { mids[0][0]-mids[3][0], mids[0][1]-mids[3][1], mids[0][2]-mids[3][2] };
  float nvw = sqrtf(vw[0]*vw[0] + vw[1]*vw[1] + vw[2]*vw[2]) + EPSF;
  for (int i = 0; i < 3; ++i) vw[i] /= nvw;
  float law = 1.f - fminf(fabsf(vw[0]*hl[0] + vw[1]*hl[1] + vw[2]*hl[2]), 1.f);
  float ppw[3];
  for (int r = 0; r < 3; ++r)
    ppw[r] = R[r*3+0]*patch_local[0] + R[r*3+1]*patch_local[1] + R[r*3+2]*patch_local[2] + palm_pos[r];
  float vf[3] = { mids[1][0]-ppw[0], mids[1][1]-ppw[1], mids[1][2]-ppw[2] };
  float nvf = sqrtf(vf[0]*vf[0] + vf[1]*vf[1] + vf[2]*vf[2]) + EPSF;
  for (int i = 0; i < 3; ++i) vf[i] /= nvf;
  float loss_align = law + fabsf(vf[0]*hl[0] + vf[1]*hl[1] + vf[2]*hl[2]);

  // ---- projections ----
  float pt[4], pm4[4];
  for (int f = 0; f < 4; ++f) {
    pt[f]  = (tips[f][0]-center[0])*ho[0] + (tips[f][1]-center[1])*ho[1] + (tips[f][2]-center[2])*ho[2];
    pm4[f] = (mids[f][0]-center[0])*ho[0] + (mids[f][1]-center[1])*ho[1] + (mids[f][2]-center[2])*ho[2];
  }
  float proj_thumb = (tips[4][0]-center[0])*ho[0] + (tips[4][1]-center[1])*ho[1] + (tips[4][2]-center[2])*ho[2];
  float mid_proj = 0.5f * (min_proj + max_proj);

  float loss_opp = fmaxf(mid_proj + 0.003f - proj_thumb, 0.f);
  for (int f = 0; f < 4; ++f) loss_opp += fmaxf(pt[f] - (mid_proj - 0.003f), 0.f);

  float db = mid_proj - 0.55f * (max_proj - min_proj + EPSF);
  float s1 = 0.f, s2 = 0.f;
  for (int f = 0; f < 4; ++f) { s1 += fabsf(pt[f] - db); s2 += fabsf(pm4[f] - db); }
  float loss_back = 0.25f * s1 + 0.5f * 0.25f * s2;

  float loss_anti = 0.f;
  for (int f = 0; f < 4; ++f) loss_anti += fmaxf(min_proj + 0.001f - pt[f], 0.f);

  // ---- flex + direction consistency ----
  float f2[4] = { qpos[2], qpos[6], qpos[14], qpos[10] };
  float f3[4] = { qpos[3], qpos[7], qpos[15], qpos[11] };
  float m2 = 0.25f*(f2[0]+f2[1]+f2[2]+f2[3]);
  float m3 = 0.25f*(f3[0]+f3[1]+f3[2]+f3[3]);
  float v2s = 0.f, v3s = 0.f;
  for (int i = 0; i < 4; ++i) { v2s += (f2[i]-m2)*(f2[i]-m2); v3s += (f3[i]-m3)*(f3[i]-m3); }
  float lfc = v2s/3.f + v3s/3.f;

  float fd[4][3], mdv[3] = {0.f, 0.f, 0.f};
  for (int f = 0; f < 4; ++f) {
    float d[3] = { tips[f][0]-mids[f][0], tips[f][1]-mids[f][1], tips[f][2]-mids[f][2] };
    float nd = sqrtf(d[0]*d[0] + d[1]*d[1] + d[2]*d[2]) + EPSF;
    for (int i = 0; i < 3; ++i) { fd[f][i] = d[i]/nd; mdv[i] += d[i]/nd; }
  }
  float nmd = sqrtf(mdv[0]*mdv[0] + mdv[1]*mdv[1] + mdv[2]*mdv[2]) + EPSF;
  for (int i = 0; i < 3; ++i) mdv[i] /= nmd;
  float ldc = 0.f;
  for (int f = 0; f < 4; ++f) ldc += 1.f - (fd[f][0]*mdv[0] + fd[f][1]*mdv[1] + fd[f][2]*mdv[2]);
  ldc *= 0.25f;
  float lc = lfc + ldc;

  // ---- thumb direction ----
  float td[3] = { tips[4][0]-mids[4][0], tips[4][1]-mids[4][1], tips[4][2]-mids[4][2] };
  float ntd = sqrtf(td[0]*td[0] + td[1]*td[1] + td[2]*td[2]) + EPSF;
  for (int i = 0; i < 3; ++i) td[i] /= ntd;
  float tpc[3];
  for (int i = 0; i < 3; ++i) tpc[i] = 0.25f*(tips[0][i]+tips[1][i]+tips[2][i]+tips[3][i]);
  float tg[3] = { tpc[0]-mids[4][0], tpc[1]-mids[4][1], tpc[2]-mids[4][2] };
  float ntg = sqrtf(tg[0]*tg[0] + tg[1]*tg[1] + tg[2]*tg[2]) + EPSF;
  for (int i = 0; i < 3; ++i) tg[i] /= ntg;
  float dth = td[0]*tg[0] + td[1]*tg[1] + td[2]*tg[2];
  dth = fminf(fmaxf(dth, -1.f), 1.f);
  float ltd = 1.f - dth;

  // ---- centerline / force closure ----
  float pl4[4], plm = 0.f;
  for (int f = 0; f < 4; ++f) {
    pl4[f] = (tips[f][0]-center[0])*hl[0] + (tips[f][1]-center[1])*hl[1] + (tips[f][2]-center[2])*hl[2];
    plm += pl4[f];
  }
  plm *= 0.25f;
  float plt = (tips[4][0]-center[0])*hl[0] + (tips[4][1]-center[1])*hl[1] + (tips[4][2]-center[2])*hl[2];
  float loss_center = fabsf(plm) + 0.5f * fabsf(plt);

  float cv[3] = { tips[4][0]-tpc[0], tips[4][1]-tpc[1], tips[4][2]-tpc[2] };
  float ncv = sqrtf(cv[0]*cv[0] + cv[1]*cv[1] + cv[2]*cv[2]) + EPSF;
  float cd[3] = { cv[0]/ncv, cv[1]/ncv, cv[2]/ncv };
  float lfa = 1.f - fminf(fabsf(cd[0]*ho[0] + cd[1]*ho[1] + cd[2]*ho[2]), 1.f);
  float ltl = fabsf(cd[0]*hl[0] + cd[1]*hl[1] + cd[2]*hl[2]);
  float lss = fabsf(plt - plm);
  float cw  = fabsf(cv[0]*ho[0] + cv[1]*ho[1] + cv[2]*ho[2]);
  float tw  = 0.9f * (max_proj - min_proj + EPSF);
  float lcw = fabsf(cw - tw);
  float lfcl = 1.2f*lfa + 0.8f*ltl + 0.6f*lss + 0.8f*lcw;

  float lna = 0.f, lnb = 0.f;
  for (int f = 1; f < 4; ++f) {
    lna += (pt[f]-pt[0])*(pt[f]-pt[0]);
    lnb += (pm4[f]-pm4[0])*(pm4[f]-pm4[0]);
  }
  lna += 0.5f * lnb;

  float st = 0.f, sm = 0.f, ltc = 0.f, lmc = 0.f;
  for (int f = 0; f < 5; ++f) {
    st += per_tip[f]; sm += per_mid[f];
    ltc += fmaxf(per_tip[f] - 0.015f, 0.f);
    lmc += fmaxf(per_mid[f] - 0.015f, 0.f);
  }
  float cham = 0.2f*st + 0.3f*0.2f*sm;

  float alpha = fminf(1.f, (float)step_p[0] / 2000.f);

  out[0] = 2.f*cham + loss_align + loss_opp + loss_back + loss_anti
         + 0.5f*lc + 0.5f*ltd + loss_center + alpha*lfcl + 0.3f*lna
         + ltc + 0.5f*lmc;
}

extern "C" void kernel_launch(void* const* d_in, const int* in_sizes, int n_in,
                              void* d_out, int out_size, void* d_ws, size_t ws_size,
                              hipStream_t stream) {
  const float* pc         = (const float*)d_in[0];
  const float* center     = (const float*)d_in[1];
  const float* hout       = (const float*)d_in[2];
  const float* hlong      = (const float*)d_in[3];
  const float* palm_pos   = (const float*)d_in[4];
  const float* palm_rot   = (const float*)d_in[5];
  const float* qpos       = (const float*)d_in[6];
  const float* tip_base   = (const float*)d_in[7];
  const float* mid_base   = (const float*)d_in[8];
  const float* J_tip      = (const float*)d_in[9];
  const float* J_mid      = (const float*)d_in[10];
  const float* patch_loc  = (const float*)d_in[12];   // palm_patch_local (palm_center_local unused)
  const int*   step_p     = (const int*)d_in[13];

  float*    wsf = (float*)d_ws;            // [0..66] WMMA constants + h_out
  unsigned* red = (unsigned*)d_ws + 80;    // [80..97] reduction cells (byte offset 320)

  long n_points = (long)in_sizes[0] / 3;
  int  n_pairs  = (int)(n_points / 32);    // 32 points (two WMMA tiles) per wave iteration

  hipLaunchKernelGGL(prep_kernel, dim3(1), dim3(32), 0, stream,
                     hout, palm_pos, palm_rot, qpos, tip_base, mid_base, J_tip, J_mid, wsf, red);
  hipLaunchKernelGGL(chamfer_kernel, dim3(2048), dim3(256), 0, stream,
                     pc, wsf, red, n_pairs);
  hipLaunchKernelGGL(finish_kernel, dim3(1), dim3(32), 0, stream,
                     pc, n_points, (long)n_pairs * 32, center, hout, hlong,
                     palm_pos, palm_rot, qpos, tip_base, mid_base, J_tip, J_mid,
                     patch_loc, step_p, red, (float*)d_out);
}